// CannyFilter_47167330844697
// MI455X (gfx1250) — compile-verified
//
#include <hip/hip_runtime.h>
#include <hip/hip_bf16.h>

typedef __attribute__((ext_vector_type(4))) int v4i;

// ---------------- CDNA5 async global->LDS helpers ----------------
__device__ __forceinline__ void async_copy_b32(const float* g, float* l) {
#if __has_builtin(__builtin_amdgcn_global_load_async_to_lds_b32)
    __builtin_amdgcn_global_load_async_to_lds_b32(
        (__attribute__((address_space(1))) int*)(unsigned long long)g,
        (__attribute__((address_space(3))) int*)l,
        /*offset=*/0, /*cpol=*/0);
#else
    unsigned loff = (unsigned)(unsigned long long)(__attribute__((address_space(3))) char*)l;
    asm volatile("global_load_async_to_lds_b32 %0, %1, off"
                 :: "v"(loff), "v"(g) : "memory");
#endif
}

__device__ __forceinline__ void async_copy_b128(const float* g, float* l) {
#if __has_builtin(__builtin_amdgcn_global_load_async_to_lds_b128)
    __builtin_amdgcn_global_load_async_to_lds_b128(
        (__attribute__((address_space(1))) v4i*)(unsigned long long)g,
        (__attribute__((address_space(3))) v4i*)l,
        /*offset=*/0, /*cpol=*/0);
#else
    async_copy_b32(g + 0, l + 0);
    async_copy_b32(g + 1, l + 1);
    async_copy_b32(g + 2, l + 2);
    async_copy_b32(g + 3, l + 3);
#endif
}

__device__ __forceinline__ void async_wait_all() {
#if __has_builtin(__builtin_amdgcn_s_wait_asynccnt)
    __builtin_amdgcn_s_wait_asynccnt(0);
#else
    asm volatile("s_wait_asynccnt 0" ::: "memory");
#endif
}

__device__ __forceinline__ int clampi(int v, int lo, int hi) {
    return v < lo ? lo : (v > hi ? hi : v);
}

// ---------------- fused Canny kernel ----------------
// Tile: 64x32 outputs per block, 256 threads (8 wave32s).
// LDS floats, aliased to stay < 64 KB static:
//   imgT  [3][38][72]  at 0      x-base tx0-4 (16B-aligned rows), y-base ty0-3
//   blurT [3][36][72]  at 8208   x-base tx0-2, y-base ty0-2
//   magT/gxT/gyT [34][68] alias imgT (dead after stage 1), base (ty0-1, tx0-1)
#define TX 64
#define TY 32
#define IH 38
#define IW 72
#define BH 36
#define BW 72
#define MH 34
#define MW 68

template <bool EDGE>
__device__ __forceinline__ void canny_tile(const float* __restrict__ imgB,
                                           float* __restrict__ outB,
                                           float* smem, const float* wG,
                                           int tx0, int ty0, int tid)
{
    const int H = 1024, W = 1024;
    float* imgT  = smem;
    float* blurT = smem + 3 * IH * IW;
    float* magT  = smem;                 // aliases imgT
    float* gxT   = smem + MH * MW;
    float* gyT   = smem + 2 * MH * MW;

    // ---- Stage 0: async-copy image tile (3 ch, halo) into LDS ----
    if (EDGE) {
        for (int i = tid; i < 3 * IH * IW; i += 256) {
            int c  = i / (IH * IW);
            int r  = (i / IW) % IH;
            int x  = i % IW;
            int py = clampi(ty0 - 3 + r, 0, H - 1);
            int px = clampi(tx0 - 4 + x, 0, W - 1);
            async_copy_b32(imgB + ((size_t)c * H + py) * W + px, imgT + i);
        }
    } else {
        // interior: 18 x b128 per 72-float row, all 16B aligned
        for (int i = tid; i < 3 * IH * 18; i += 256) {
            int c = i / (IH * 18);
            int r = (i / 18) % IH;
            int q = i % 18;
            const float* src = imgB + ((size_t)c * H + (ty0 - 3 + r)) * W + (tx0 - 4) + q * 4;
            async_copy_b128(src, imgT + (c * IH + r) * IW + q * 4);
        }
    }
    async_wait_all();
    __syncthreads();

    // ---- Stage 1: Gaussian blur ----
    for (int i = tid; i < 3 * BH * 68; i += 256) {
        int c = i / (BH * 68);
        int r = (i / 68) % BH;
        int x = i % 68;
        const float* P = imgT + c * IH * IW;
        float s;
        if (!EDGE) {
            int o = r * IW + (x + 1);
            s = wG[0]*P[o]        + wG[1]*P[o+1]        + wG[2]*P[o+2]
              + wG[3]*P[o+IW]     + wG[4]*P[o+IW+1]     + wG[5]*P[o+IW+2]
              + wG[6]*P[o+2*IW]   + wG[7]*P[o+2*IW+1]   + wG[8]*P[o+2*IW+2];
        } else {
            int cy = clampi(ty0 - 2 + r, 0, H - 1);   // stored = blur at clamped pos
            int cx = clampi(tx0 - 2 + x, 0, W - 1);
            int r0 = (cy - 1 < 0     ? 0     : cy - 1) - (ty0 - 3);
            int r1 =  cy                                - (ty0 - 3);
            int r2 = (cy + 1 > H - 1 ? H - 1 : cy + 1) - (ty0 - 3);
            int x0 = (cx - 1 < 0     ? 0     : cx - 1) - (tx0 - 4);
            int x1 =  cx                                - (tx0 - 4);
            int x2 = (cx + 1 > W - 1 ? W - 1 : cx + 1) - (tx0 - 4);
            s = wG[0]*P[r0*IW+x0] + wG[1]*P[r0*IW+x1] + wG[2]*P[r0*IW+x2]
              + wG[3]*P[r1*IW+x0] + wG[4]*P[r1*IW+x1] + wG[5]*P[r1*IW+x2]
              + wG[6]*P[r2*IW+x0] + wG[7]*P[r2*IW+x1] + wG[8]*P[r2*IW+x2];
        }
        blurT[(c * BH + r) * BW + x] = s;
    }
    __syncthreads();

    // ---- Stage 2: Sobel (exact +-0.5/+-1 taps), channel mean, mag = gx^2+gy^2 ----
    for (int i = tid; i < MH * 66; i += 256) {
        int r = i / 66;
        int x = i % 66;
        int r0, r1, r2, x0, x1, x2;
        if (!EDGE) {
            r0 = r; r1 = r + 1; r2 = r + 2;
            x0 = x; x1 = x + 1; x2 = x + 2;
        } else {
            int cy = clampi(ty0 - 1 + r, 0, H - 1);
            int cx = clampi(tx0 - 1 + x, 0, W - 1);
            r0 = (cy - 1 < 0     ? 0     : cy - 1) - (ty0 - 2);
            r1 =  cy                                - (ty0 - 2);
            r2 = (cy + 1 > H - 1 ? H - 1 : cy + 1) - (ty0 - 2);
            x0 = (cx - 1 < 0     ? 0     : cx - 1) - (tx0 - 2);
            x1 =  cx                                - (tx0 - 2);
            x2 = (cx + 1 > W - 1 ? W - 1 : cx + 1) - (tx0 - 2);
        }
        float sgx = 0.0f, sgy = 0.0f;
        #pragma unroll
        for (int c = 0; c < 3; ++c) {
            const float* Bp = blurT + c * BH * BW;
            float tl = Bp[r0*BW+x0], tm = Bp[r0*BW+x1], tr = Bp[r0*BW+x2];
            float ml = Bp[r1*BW+x0],                     mr = Bp[r1*BW+x2];
            float bl = Bp[r2*BW+x0], bm = Bp[r2*BW+x1], br = Bp[r2*BW+x2];
            sgx += 0.5f*(tr - tl) + (mr - ml) + 0.5f*(br - bl);
            sgy += 0.5f*(bl - tl) + (bm - tm) + 0.5f*(br - tr);
        }
        float gxv = sgx * (1.0f / 3.0f);
        float gyv = sgy * (1.0f / 3.0f);
        int o = r * MW + x;
        magT[o] = gxv * gxv + gyv * gyv;
        gxT[o]  = gxv;
        gyT[o]  = gyv;
    }
    __syncthreads();

    // ---- Stage 3: orientation (only idx 3/7 matter), NMS vs diagonals, store ----
    for (int i = tid; i < TY * TX; i += 256) {
        int r  = i / TX;
        int x  = i % TX;
        int py = ty0 + r;
        int px = tx0 + x;
        int mc = (r + 1) * MW + (x + 1);
        float m = magT[mc];
        float mUL, mDR;
        if (!EDGE) {
            mUL = magT[r * MW + x];
            mDR = magT[(r + 2) * MW + (x + 2)];
        } else {
            int uy = (py - 1 < 0     ? 0     : py - 1) - (ty0 - 1);
            int ux = (px - 1 < 0     ? 0     : px - 1) - (tx0 - 1);
            int dy = (py + 1 > H - 1 ? H - 1 : py + 1) - (ty0 - 1);
            int dx = (px + 1 > W - 1 ? W - 1 : px + 1) - (tx0 - 1);
            mUL = magT[uy * MW + ux];
            mDR = magT[dy * MW + dx];
        }
        float d3 = m - mUL;                 // dir kernel 3: center - up-left
        float d7 = m - mDR;                 // dir kernel 7: center - down-right
        bool isMax = fminf(d3, d7) > 0.0f;
        float gxv = gxT[mc], gyv = gyT[mc];
        // ori/45 = atan(gy/gx)*8/pi + 4 ; jnp.round == rintf (half-to-even)
        float t = __builtin_rintf(atanf(gyv / gxv) * 2.546479089470325f + 4.0f);
        bool oriented = (t == 3.0f) || (t == 7.0f);   // NaN -> false, t==8 -> idx 0
        outB[(size_t)py * W + px] = (oriented && !isMax) ? 0.0f : m;
    }
}

__global__ __launch_bounds__(256)
void canny_fused_kernel(const float* __restrict__ img,
                        const float* __restrict__ gauss,
                        float* __restrict__ out)
{
    __shared__ __align__(16) float smem[3*IH*IW + 3*BH*BW];  // 15984 floats = 63936 B
    __shared__ float wG[9];

    const int tid = threadIdx.x;
    const int bx  = blockIdx.x, by = blockIdx.y, b = blockIdx.z;
    const int tx0 = bx * TX;
    const int ty0 = by * TY;

    if (tid < 9) wG[tid] = gauss[tid];

    const float* imgB = img + (size_t)b * 3 * 1024 * 1024;
    float*       outB = out + (size_t)b * 1024 * 1024;

    // Block-uniform branch (blockIdx only) -> barriers stay uniform.
    // Interior: whole 7x7 dependency footprint in-bounds, no clamping anywhere.
    const bool interior = (bx >= 1) && (bx <= 14) && (by >= 1) && (by <= 30);
    if (interior) canny_tile<false>(imgB, outB, smem, wG, tx0, ty0, tid);
    else          canny_tile<true >(imgB, outB, smem, wG, tx0, ty0, tid);
}

extern "C" void kernel_launch(void* const* d_in, const int* in_sizes, int n_in,
                              void* d_out, int out_size, void* d_ws, size_t ws_size,
                              hipStream_t stream) {
    (void)in_sizes; (void)n_in; (void)out_size; (void)d_ws; (void)ws_size;
    const float* img   = (const float*)d_in[0];   // (8,3,1024,1024) f32
    const float* gauss = (const float*)d_in[1];   // (1,1,3,3) f32
    float* out = (float*)d_out;                   // (8,1,1024,1024) f32
    dim3 grid(1024 / TX, 1024 / TY, 8);
    canny_fused_kernel<<<grid, dim3(256, 1, 1), 0, stream>>>(img, gauss, out);
}